// _CompiledDemographyStep_67233418052194
// MI455X (gfx1250) — compile-verified
//
#include <hip/hip_runtime.h>
#include <math.h>

typedef float v2f __attribute__((ext_vector_type(2)));
typedef float v8f __attribute__((ext_vector_type(8)));
typedef int   b128v __attribute__((vector_size(16)));

namespace {

constexpr int   R_  = 4096;
constexpr int   A_  = 101;
constexpr int   AK  = 104;   // K padded to multiple of 4
constexpr int   AN  = 112;   // N padded to multiple of 16
constexpr int   K_  = 32;
constexpr int   N_  = 4000000;
constexpr float EPS = 1e-9f;
constexpr float DT  = 1.0f / 365.0f;
constexpr float FRAC_MOVE = (float)(0.02 / 365.0);

// ---- d_ws float offsets ----
constexpr size_t W_SURV  = 0;                        // R*AK (K-padded survivors, A matrix)
constexpr size_t W_POPA  = W_SURV + (size_t)R_*AK;   // R*AN (pop_after, padded)
constexpr size_t W_BPAD  = W_POPA + (size_t)R_*AN;   // AK*AN (padded aging_M)
constexpr size_t W_ATTR  = W_BPAD + (size_t)AK*AN;   // R
constexpr size_t W_FDC   = W_ATTR + R_;              // R  (F_dev * F_cyc)
constexpr size_t W_H     = W_FDC  + R_;              // R
constexpr size_t W_UTIL  = W_H    + R_;              // R
constexpr size_t W_FTOT  = W_UTIL + R_;              // R
constexpr size_t W_DFRAC = W_FTOT + R_;              // R
constexpr size_t W_HAZ0  = W_DFRAC+ R_;              // R
constexpr size_t W_BSURV = W_HAZ0 + R_;              // R
constexpr size_t W_SNAP  = W_BSURV+ R_;              // R*22 mobile snapshot
constexpr size_t W_MSUM  = W_SNAP + (size_t)R_*22;   // R
constexpr size_t W_WSUM  = W_MSUM + R_;              // R  -- atomic accumulators start here
constexpr size_t W_SATP  = W_WSUM + R_;              // R
constexpr size_t W_SADP  = W_SATP + R_;              // R
constexpr size_t W_SAMP  = W_SADP + R_;              // R
constexpr size_t W_CNT   = W_SAMP + R_;              // R
constexpr size_t W_HEIRA = W_CNT  + R_;              // R
constexpr size_t W_HEIRD = W_HEIRA+ R_;              // R
constexpr size_t W_HEIRM = W_HEIRD+ R_;              // R
constexpr size_t W_ENDOW = W_HEIRM+ R_;              // R
constexpr size_t W_SCAL  = W_ENDOW+ R_;              // [0]=attract_sum [1]=dist_sum
constexpr size_t W_TOTAL = W_SCAL + 8;

// ---- d_out float offsets ----
constexpr size_t O_POP   = 0;                        // R*A
constexpr size_t O_POPN  = O_POP + (size_t)R_*A_;    // R
constexpr size_t O_GDP   = O_POPN + R_;
constexpr size_t O_DEV   = O_GDP  + R_;
constexpr size_t O_H     = O_DEV  + R_;
constexpr size_t O_CONS  = O_H    + R_;
constexpr size_t O_LAB   = O_CONS + R_;
constexpr size_t O_PSR   = O_LAB  + R_;
constexpr size_t O_DEP   = O_PSR  + R_;
constexpr size_t O_ATP   = O_DEP  + R_;              // N
constexpr size_t O_ADP   = O_ATP  + (size_t)N_;      // N
constexpr size_t O_AMP   = O_ADP  + (size_t)N_;      // N
constexpr size_t O_PLA   = O_AMP  + (size_t)N_;      // R
constexpr size_t O_PLD   = O_PLA  + R_;
constexpr size_t O_PLM   = O_PLD  + R_;

__device__ __forceinline__ float clampf(float x, float lo, float hi) {
  return fminf(fmaxf(x, lo), hi);
}

// ---------------- kernels ----------------

__global__ void k_zero(float* ws) {
  size_t i = (size_t)blockIdx.x * blockDim.x + threadIdx.x;
  size_t n = W_TOTAL - W_WSUM;
  if (i < n) ws[W_WSUM + i] = 0.0f;
}

__global__ void k_region_pre(const float* __restrict__ aec, const float* __restrict__ gdp,
                             const float* __restrict__ ema_in, const float* __restrict__ base,
                             const float* __restrict__ devin, const float* __restrict__ suse,
                             const float* __restrict__ scap, float* __restrict__ out,
                             float* __restrict__ ws) {
  int r = blockIdx.x * blockDim.x + threadIdx.x;
  if (r >= R_) return;
  float aec_norm = clampf(aec[r], 0.f, 1.f);
  float prev = ema_in[r];
  float ge = prev * 0.9f + 0.1f * gdp[r];
  float g_ratio = logf(fmaxf(ge / (base[r] + EPS), 1e-6f));
  float g_term = clampf(0.5f + 0.5f * tanhf(0.5f * g_ratio), 0.f, 1.f);
  float util = clampf(suse[r] / (scap[r] + EPS), 0.f, 1.f);
  float H = clampf(0.5f * aec_norm + 0.4f * g_term + 0.1f * (1.f - util), 0.f, 1.f);
  float dev_proxy = clampf(0.5f + 0.5f * tanhf(0.3f * g_ratio), 0.f, 1.f);
  float dev = devin[r] * 0.99f + 0.01f * dev_proxy;
  float F_dev = clampf(expf(-dev), 0.5f, 1.5f);
  float g_growth = logf(fmaxf(ge + EPS, 1e-6f)) - logf(fmaxf(prev + EPS, 1e-6f));
  float F_cyc = clampf(expf(-5.f * fmaxf(-g_growth, 0.f)), 0.6f, 1.2f);
  float attract = 0.6f * (gdp[r] / (base[r] + EPS)) + 0.4f * (0.5f + 0.5f * aec_norm);
  out[O_GDP + r] = ge;
  out[O_DEV + r] = dev;
  out[O_H + r]   = H;
  ws[W_ATTR + r] = attract;
  ws[W_FDC + r]  = F_dev * F_cyc;
  ws[W_H + r]    = H;
  ws[W_UTIL + r] = util;
}

__global__ void k_attr_reduce(float* __restrict__ ws) {
  __shared__ float sh[1024];
  int t = threadIdx.x;
  float s = 0.f;
  for (int i = t; i < R_; i += 1024) s += ws[W_ATTR + i];
  sh[t] = s; __syncthreads();
  for (int o = 512; o > 0; o >>= 1) { if (t < o) sh[t] += sh[t + o]; __syncthreads(); }
  if (t == 0) ws[W_SCAL + 0] = sh[0];
}

__global__ void k_dist_reduce(const float* __restrict__ distance, const int* __restrict__ nbr,
                              float* __restrict__ ws) {
  __shared__ float sh[256];
  int i = blockIdx.x * blockDim.x + threadIdx.x;
  float v = 0.f;
  if (i < R_ * K_) {
    int r = i / K_;
    int nb = nbr[i];
    v = distance[(size_t)r * R_ + nb];
  }
  int t = threadIdx.x;
  sh[t] = v; __syncthreads();
  for (int o = 128; o > 0; o >>= 1) { if (t < o) sh[t] += sh[t + o]; __syncthreads(); }
  if (t == 0) atomicAdd(&ws[W_SCAL + 1], sh[0]);
}

__global__ void k_survivors(const float* __restrict__ hbase, const float* __restrict__ pop_age,
                            const float* __restrict__ population, float* __restrict__ ws) {
  __shared__ float sp[128], ss[128], s5[128];
  int r = blockIdx.x, a = threadIdx.x;
  float H = ws[W_H + r], util = ws[W_UTIL + r];
  float p = 0.f, sv = 0.f, h5 = 0.f;
  if (a < A_) {
    float mult = (a == 0) ? expf(-2.0f * H) : ((a < 15) ? expf(-1.5f * H) : expf(-H));
    float h = clampf(hbase[a] * mult * (1.f + 0.5f * util), 0.f, 5.f);
    float S = expf(-h * DT);
    p = pop_age[(size_t)r * A_ + a];
    sv = p * S;
    ws[W_SURV + (size_t)r * AK + a] = sv;
    if (a < 5) h5 = h;
    if (a == 0) ws[W_HAZ0 + r] = h;
  } else if (a < AK) {
    ws[W_SURV + (size_t)r * AK + a] = 0.f;  // K padding
  }
  sp[a] = p; ss[a] = sv; s5[a] = h5;
  __syncthreads();
  for (int o = 64; o > 0; o >>= 1) {
    if (a < o) { sp[a] += sp[a + o]; ss[a] += ss[a + o]; s5[a] += s5[a + o]; }
    __syncthreads();
  }
  if (a == 0) {
    float deaths = fmaxf(sp[0] - ss[0], 0.f);
    float dfrac = clampf(deaths / (population[r] + EPS), 0.f, 0.99f);
    float su5 = expf(-s5[0]);
    float F_rep = clampf(0.995f / fmaxf(su5, 0.001f), 0.5f, 1.8f);
    ws[W_FTOT + r]  = clampf(ws[W_FDC + r] * F_rep, 0.4f, 1.8f);
    ws[W_DFRAC + r] = dfrac;
  }
}

__global__ void k_padB(const float* __restrict__ agingM, float* __restrict__ ws) {
  int i = blockIdx.x * blockDim.x + threadIdx.x;
  if (i >= AK * AN) return;
  int kk = i / AN, nn = i % AN;
  ws[W_BPAD + i] = (kk < A_ && nn < A_) ? agingM[kk * A_ + nn] : 0.f;
}

// pop_after_pad = surv_pad (4096 x 104) @ Bpad (104 x 112) via V_WMMA_F32_16X16X4_F32.
// Bpad (46.6 KB) is shared by every M-tile: stage it once per workgroup into LDS
// through the gfx1250 async load-to-LDS path (ASYNCcnt), then feed B fragments
// from LDS (ds_load) while A fragments stream from global.
__global__ void k_wmma_gemm(float* __restrict__ ws) {
  __shared__ float shB[AK * AN];
  const int NT = AN / 16;  // 7 tiles along N
  float* Bg = ws + W_BPAD;
  // cooperative async copy: 104*112*4 B = 2912 x b128
  for (int c = threadIdx.x; c < (AK * AN) / 4; c += blockDim.x) {
    __builtin_amdgcn_global_load_async_to_lds_b128(
        (__attribute__((address_space(1))) b128v*)(Bg + 4 * c),
        (__attribute__((address_space(3))) b128v*)(shB + 4 * c),
        0, 0);
  }
  __builtin_amdgcn_s_wait_asynccnt(0);
  __syncthreads();

  // grid is sized exactly: 224 blocks * 8 waves == 1792 tiles == (R/16)*(AN/16)
  int tile = blockIdx.x * (blockDim.x >> 5) + (threadIdx.x >> 5);
  int lane = threadIdx.x & 31;
  int mt = tile / NT, nt = tile % NT;
  int half = lane >> 4;                       // 0 or 1
  int row  = mt * 16 + (lane & 15);           // A: lane selects M
  int colg = nt * 16 + (lane & 15);           // B/C: lane selects N
  int coll = nt * 16 + (lane & 15);           // column inside shB
  const float* __restrict__ Ap = ws + W_SURV;
  float* __restrict__ Cp = ws + W_POPA;
  v8f acc = {};
  for (int k0 = 0; k0 < AK; k0 += 4) {
    v2f a, b;
    const float* ap = Ap + (size_t)row * AK + k0 + 2 * half;
    a.x = ap[0];
    a.y = ap[1];
    b.x = shB[(k0 + 2 * half + 0) * AN + coll];
    b.y = shB[(k0 + 2 * half + 1) * AN + coll];
    acc = __builtin_amdgcn_wmma_f32_16x16x4_f32(false, a, false, b,
                                                (short)0, acc, false, false);
  }
#pragma unroll
  for (int v = 0; v < 8; ++v) {
    int m = mt * 16 + v + 8 * half;
    Cp[(size_t)m * AN + colg] = acc[v];
  }
}

__global__ void k_births(const float* __restrict__ asfr, float* __restrict__ ws) {
  __shared__ float sb[64];
  __shared__ float s_births;
  int r = blockIdx.x, t = threadIdx.x;
  float* popA = ws + W_POPA + (size_t)r * AN;
  float v = 0.f;
  if (t < 35) v = 0.5f * popA[15 + t] * asfr[t];
  sb[t] = v; __syncthreads();
  for (int o = 32; o > 0; o >>= 1) { if (t < o) sb[t] += sb[t + o]; __syncthreads(); }
  if (t == 0) {
    float births = fmaxf(sb[0] * ws[W_FTOT + r] * DT, 0.f);
    float bs = births * expf(-ws[W_HAZ0 + r] * DT);
    popA[0] += bs;
    ws[W_BSURV + r] = bs;
    s_births = bs;
  }
  __syncthreads();
  (void)s_births;
  // snapshot mobile slice (ages 18..39) before migration atomics run
  float m = 0.f;
  if (t < 22) { m = popA[18 + t]; ws[W_SNAP + (size_t)r * 22 + t] = m; }
  sb[t] = m; __syncthreads();
  for (int o = 32; o > 0; o >>= 1) { if (t < o) sb[t] += sb[t + o]; __syncthreads(); }
  if (t == 0) ws[W_MSUM + r] = sb[0];
}

__global__ void k_migration(const float* __restrict__ distance, const int* __restrict__ nbr,
                            float* __restrict__ ws) {
  int wave = threadIdx.x >> 5, lane = threadIdx.x & 31;
  int r = blockIdx.x * (blockDim.x >> 5) + wave;
  if (r >= R_) return;
  float am = ws[W_SCAL + 0] / (float)R_;
  float dm = ws[W_SCAL + 1] / (float)(R_ * K_);
  int nb = nbr[(size_t)r * K_ + lane];
  float dist = distance[(size_t)r * R_ + nb];
  float cost = 1.f + dist / (dm + EPS);
  float an = ws[W_ATTR + nb] / (am + EPS);
  float wr = fmaxf(an / cost, 0.f);          // KAPPA == 1
  float wsum = wr;
#pragma unroll
  for (int m = 16; m > 0; m >>= 1) wsum += __shfl_xor(wsum, m, 32);
  float msum = ws[W_MSUM + r];
  float outR = msum * FRAC_MOVE;
  float wn = wr / (wsum + EPS);
  float move = outR * wn;
  float move_tot = outR * (wsum / (wsum + EPS));  // outR * sum(w_norm)
  float inv = 1.f / (msum + EPS);
  if (lane < 22) {
    float share = ws[W_SNAP + (size_t)r * 22 + lane] * inv;
    atomicAdd(&ws[W_POPA + (size_t)r * AN + 18 + lane], -share * move_tot);
  }
  float* dst = &ws[W_POPA + (size_t)nb * AN + 18];
#pragma unroll
  for (int a2 = 0; a2 < 22; ++a2) {
    float share = ws[W_SNAP + (size_t)r * 22 + a2] * inv;
    atomicAdd(&dst[a2], share * move);
  }
}

__global__ void k_finalize(const float* __restrict__ cw, const float* __restrict__ pw,
                           const float* __restrict__ cbase, const float* __restrict__ lbase,
                           float* __restrict__ out, const float* __restrict__ ws) {
  __shared__ float s0[128], s1[128], s2[128], s3[128], s4[128], s5[128];
  int r = blockIdx.x, a = threadIdx.x;
  float p = 0.f, c = 0.f, l = 0.f, wk = 0.f, yg = 0.f, od = 0.f;
  if (a < A_) {
    p = fmaxf(ws[W_POPA + (size_t)r * AN + a], 0.f);
    out[O_POP + (size_t)r * A_ + a] = p;
    c = p * cw[a];
    l = p * pw[a];
    if (a >= 15 && a < 65) wk = p;
    if (a < 15) yg = p;
    if (a >= 65) od = p;
  }
  s0[a] = p; s1[a] = c; s2[a] = l; s3[a] = wk; s4[a] = yg; s5[a] = od;
  __syncthreads();
  for (int o = 64; o > 0; o >>= 1) {
    if (a < o) {
      s0[a] += s0[a + o]; s1[a] += s1[a + o]; s2[a] += s2[a + o];
      s3[a] += s3[a + o]; s4[a] += s4[a + o]; s5[a] += s5[a + o];
    }
    __syncthreads();
  }
  if (a == 0) {
    out[O_POPN + r] = fmaxf(s0[0], 0.f);
    out[O_CONS + r] = clampf(s1[0] / (cbase[r] + EPS), 0.25f, 4.f);
    out[O_LAB + r]  = clampf(s2[0] / (lbase[r] + EPS), 0.2f, 1.2f);
    out[O_PSR + r]  = s3[0] / (s5[0] + EPS);
    out[O_DEP + r]  = (s4[0] + s5[0]) / (s3[0] + EPS);
  }
}

// LDS-binned segment sums over 4M agents: greed-weights, eATP, eADP (phase 1),
// then eAMP + counts (phase 2).  48 KB static LDS.
__global__ __launch_bounds__(1024) void k_agents_pass1(
    const float* __restrict__ greed, const float* __restrict__ eATP,
    const float* __restrict__ eADP, const float* __restrict__ eAMP,
    const int* __restrict__ areg, float* __restrict__ ws) {
  __shared__ float b0[R_], b1[R_], b2[R_];
  int t = threadIdx.x, T = blockDim.x;
  for (int i = t; i < R_; i += T) { b0[i] = 0.f; b1[i] = 0.f; b2[i] = 0.f; }
  __syncthreads();
  int stride = gridDim.x * T;
  for (int i = blockIdx.x * T + t; i < N_; i += stride) {
    int r = areg[i];
    atomicAdd(&b0[r], greed[i] + 1e-9f);   // INHERIT_CONC == 1
    atomicAdd(&b1[r], eATP[i]);
    atomicAdd(&b2[r], eADP[i]);
  }
  __syncthreads();
  for (int i = t; i < R_; i += T) {
    atomicAdd(&ws[W_WSUM + i], b0[i]);
    atomicAdd(&ws[W_SATP + i], b1[i]);
    atomicAdd(&ws[W_SADP + i], b2[i]);
    b0[i] = 0.f; b1[i] = 0.f;
  }
  __syncthreads();
  for (int i = blockIdx.x * T + t; i < N_; i += stride) {
    int r = areg[i];
    atomicAdd(&b0[r], eAMP[i]);
    atomicAdd(&b1[r], 1.0f);
  }
  __syncthreads();
  for (int i = t; i < R_; i += T) {
    atomicAdd(&ws[W_SAMP + i], b0[i]);
    atomicAdd(&ws[W_CNT + i], b1[i]);
  }
}

__global__ void k_pools(const float* __restrict__ poolA, const float* __restrict__ poolD,
                        const float* __restrict__ poolM, float* __restrict__ out,
                        float* __restrict__ ws) {
  int r = blockIdx.x * blockDim.x + threadIdx.x;
  if (r >= R_) return;
  float dfi = ws[W_DFRAC + r];
  float rpA = dfi * ws[W_SATP + r];
  float rpD = dfi * ws[W_SADP + r];
  float rpM = dfi * ws[W_SAMP + r];
  ws[W_HEIRA + r] = rpA * 0.8f;
  ws[W_HEIRD + r] = rpD * 0.8f;
  ws[W_HEIRM + r] = rpM * 0.8f;
  out[O_PLA + r] = poolA[r] - (rpA - rpA * 0.8f);
  out[O_PLD + r] = poolD[r] - (rpD - rpD * 0.8f);
  out[O_PLM + r] = poolM[r] - (rpM - rpM * 0.8f);
  ws[W_ENDOW + r] = 1.0f * (ws[W_BSURV + r] / fmaxf(ws[W_CNT + r], 1.0f));  // BIRTH_ENDOW=1
}

__global__ void k_agents_pass2(const float* __restrict__ greed, const float* __restrict__ eATP,
                               const float* __restrict__ eADP, const float* __restrict__ eAMP,
                               const int* __restrict__ areg, float* __restrict__ out,
                               const float* __restrict__ ws) {
  int i = blockIdx.x * blockDim.x + threadIdx.x;
  if (i >= N_) return;
  int r = areg[i];
  float dfi = ws[W_DFRAC + r];
  float wn = (greed[i] + 1e-9f) / (ws[W_WSUM + r] + EPS);
  float keep = 1.f - dfi;
  out[O_ATP + i] = eATP[i] * keep + wn * ws[W_HEIRA + r] + ws[W_ENDOW + r];
  out[O_ADP + i] = eADP[i] * keep + wn * ws[W_HEIRD + r];
  out[O_AMP + i] = eAMP[i] * keep + wn * ws[W_HEIRM + r];
}

}  // namespace

extern "C" void kernel_launch(void* const* d_in, const int* in_sizes, int n_in,
                              void* d_out, int out_size, void* d_ws, size_t ws_size,
                              hipStream_t stream) {
  (void)in_sizes; (void)n_in; (void)out_size; (void)ws_size;
  const float* aec       = (const float*)d_in[0];
  const float* gdp       = (const float*)d_in[1];
  const float* ema       = (const float*)d_in[2];
  const float* base      = (const float*)d_in[3];
  const float* devi      = (const float*)d_in[4];
  const float* suse      = (const float*)d_in[5];
  const float* scap      = (const float*)d_in[6];
  const float* hbase     = (const float*)d_in[7];
  const float* pop_age   = (const float*)d_in[8];
  const float* popln     = (const float*)d_in[9];
  const float* agingM    = (const float*)d_in[10];
  const float* asfr      = (const float*)d_in[11];
  const float* distance  = (const float*)d_in[12];
  const float* cw        = (const float*)d_in[13];
  const float* pw        = (const float*)d_in[14];
  const float* cbase     = (const float*)d_in[15];
  const float* lbase     = (const float*)d_in[16];
  const float* greed     = (const float*)d_in[17];
  const float* eATP      = (const float*)d_in[18];
  const float* eADP      = (const float*)d_in[19];
  const float* eAMP      = (const float*)d_in[20];
  const float* poolA     = (const float*)d_in[21];
  const float* poolD     = (const float*)d_in[22];
  const float* poolM     = (const float*)d_in[23];
  const int*   nbr       = (const int*)d_in[24];
  const int*   areg      = (const int*)d_in[25];
  float* out = (float*)d_out;
  float* ws  = (float*)d_ws;

  {
    int n = (int)(W_TOTAL - W_WSUM);
    k_zero<<<(n + 255) / 256, 256, 0, stream>>>(ws);
  }
  k_region_pre<<<(R_ + 255) / 256, 256, 0, stream>>>(aec, gdp, ema, base, devi, suse, scap,
                                                     out, ws);
  k_attr_reduce<<<1, 1024, 0, stream>>>(ws);
  k_dist_reduce<<<(R_ * K_ + 255) / 256, 256, 0, stream>>>(distance, nbr, ws);
  k_survivors<<<R_, 128, 0, stream>>>(hbase, pop_age, popln, ws);
  k_padB<<<(AK * AN + 255) / 256, 256, 0, stream>>>(agingM, ws);
  {
    int tiles = (R_ / 16) * (AN / 16);   // 256 * 7 = 1792 waves
    int waves_per_block = 8;             // 256 threads
    k_wmma_gemm<<<(tiles + waves_per_block - 1) / waves_per_block, 32 * waves_per_block, 0,
                  stream>>>(ws);
  }
  k_births<<<R_, 64, 0, stream>>>(asfr, ws);
  k_migration<<<R_ / 8, 256, 0, stream>>>(distance, nbr, ws);
  k_finalize<<<R_, 128, 0, stream>>>(cw, pw, cbase, lbase, out, ws);
  k_agents_pass1<<<256, 1024, 0, stream>>>(greed, eATP, eADP, eAMP, areg, ws);
  k_pools<<<(R_ + 255) / 256, 256, 0, stream>>>(poolA, poolD, poolM, out, ws);
  k_agents_pass2<<<(N_ + 255) / 256, 256, 0, stream>>>(greed, eATP, eADP, eAMP, areg, out, ws);
}